// S6Layer_68736656605811
// MI455X (gfx1250) — compile-verified
//
#include <hip/hip_runtime.h>
#include <math.h>

// ---------------------------------------------------------------------------
// S6 / Mamba selective scan, MI455X (gfx1250, wave32).
//   Phase 1 (s6_proj):  WMMA f32 16x16x4 GEMMs (verified: v_wmma_f32_16x16x4_f32)
//       dt_low = x @ W_dt, x_bc = x @ W_bc, delta = softplus(dt_low @ W_dtproj + b)
//     dt_low chained GEMM1->GEMM2 through per-wave LDS tile (no HBM trip).
//   Phase 2 (s6_scan):  chunked scan (8 waves = 8 L-chunks per (b,d)-pair block,
//     LDS carry propagation). Latency-bound, so data movement is taken off the
//     serial critical path with GLOBAL_LOAD_ASYNC_TO_LDS double-buffered tiles
//     (ASYNCcnt-gated): 16 timesteps of x_bc rows + delta/x strided columns per
//     tile; the recurrence then reads only LDS.
// ---------------------------------------------------------------------------

namespace {
constexpr int kB = 2;
constexpr int kL = 2048;
constexpr int kD = 1024;
constexpr int kN = 16;
constexpr int kR = 64;             // DT_RANK
constexpr int kM = kB * kL;        // 4096 flattened (b,l) rows
constexpr int kChunks = 8;
constexpr int kLC = kL / kChunks;  // 256 timesteps per chunk
constexpr int kTL = 16;            // timesteps per async tile
constexpr int kNT = kLC / kTL;     // 16 tiles per chunk
}

typedef __attribute__((ext_vector_type(2))) float v2f;
typedef __attribute__((ext_vector_type(8))) float v8f;
typedef __attribute__((ext_vector_type(2))) int   v2i;
typedef __attribute__((ext_vector_type(4))) int   v4i;

#define AS_GLOBAL __attribute__((address_space(1)))
#define AS_LDS    __attribute__((address_space(3)))

#if defined(__gfx1250__) &&                                                  \
    __has_builtin(__builtin_amdgcn_global_load_async_to_lds_b128) &&         \
    __has_builtin(__builtin_amdgcn_global_load_async_to_lds_b64) &&          \
    __has_builtin(__builtin_amdgcn_s_wait_asynccnt)
#define HAVE_ASYNC_LDS 1
#else
#define HAVE_ASYNC_LDS 0
#endif

#if HAVE_ASYNC_LDS
#define WAIT_ASYNC(n) __builtin_amdgcn_s_wait_asynccnt(n)
#else
#define WAIT_ASYNC(n)
#endif

__device__ __forceinline__ void async_b128(const float* g, float* l) {
#if HAVE_ASYNC_LDS
  __builtin_amdgcn_global_load_async_to_lds_b128((AS_GLOBAL v4i*)g,
                                                 (AS_LDS v4i*)l, 0, 0);
#else
  *(float4*)l = *(const float4*)g;
#endif
}

__device__ __forceinline__ void async_b64(const float* g, float* l) {
#if HAVE_ASYNC_LDS
  __builtin_amdgcn_global_load_async_to_lds_b64((AS_GLOBAL v2i*)g,
                                                (AS_LDS v2i*)l, 0, 0);
#else
  *(double*)l = *(const double*)g;  // bitwise 8-byte copy
#endif
}

// D = A(16x4,f32) * B(4x16,f32) + C(16x16,f32); wave32, one matrix per wave.
__device__ __forceinline__ v8f wmma4f32(v2f a, v2f b, v8f c) {
  return __builtin_amdgcn_wmma_f32_16x16x4_f32(false, a, false, b, (short)0, c,
                                               false, false);
}

__device__ __forceinline__ float softplus_f(float v) {
  return (v > 20.0f) ? v : log1pf(__expf(v));
}

// ---------------------------------------------------------------------------
// Phase 1: projections. One wave owns a 16-row (b,l) tile.
// A-fragment (16x4 f32): lanes 0-15 -> M=lane, K = k0 + 2*half + {0,1}.
// B-fragment (4x16 f32): lanes 0-15 -> N=lane, K = k0 + 2*half + {0,1}.
// C/D (16x16 f32): VGPR r: lanes 0-15 -> M=r, lanes 16-31 -> M=r+8.
// ---------------------------------------------------------------------------
__global__ __launch_bounds__(64) void s6_proj(
    const float* __restrict__ x,
    const float* __restrict__ W_bc,       // (D, 32)
    const float* __restrict__ W_dt,       // (D, 64)
    const float* __restrict__ W_dtproj,   // (64, D)
    const float* __restrict__ b_dtproj,   // (D)
    float* __restrict__ delta_ws,         // (M, D)
    float* __restrict__ xbc_ws)           // (M, 32)
{
  __shared__ float lds_dt[2][16][68];  // per-wave 16x64 dt_low tile (+4 pad)

  const int w    = threadIdx.x >> 5;
  const int lane = threadIdx.x & 31;
  const int half = lane >> 4;   // K sub-pair select
  const int lr   = lane & 15;   // A row / B,C col
  const int m0   = (blockIdx.x * 2 + w) * 16;

  const v8f zero8 = {0.f, 0.f, 0.f, 0.f, 0.f, 0.f, 0.f, 0.f};
  v8f acc_dt[4] = {zero8, zero8, zero8, zero8};  // dt_low cols 0..63
  v8f acc_bc[2] = {zero8, zero8};                // x_bc  cols 0..31

  const float* xrow = x + (size_t)(m0 + lr) * kD;

  // GEMM1: K = D = 1024, step 4. A shared across all 6 column tiles.
  for (int k0 = 0; k0 < kD; k0 += 4) {
    const int ka = k0 + 2 * half;
    v2f a;
    a.x = xrow[ka];
    a.y = xrow[ka + 1];
#pragma unroll
    for (int t = 0; t < 4; ++t) {
      v2f bf;
      bf.x = W_dt[ka * kR + t * 16 + lr];
      bf.y = W_dt[(ka + 1) * kR + t * 16 + lr];
      acc_dt[t] = wmma4f32(a, bf, acc_dt[t]);
    }
#pragma unroll
    for (int t = 0; t < 2; ++t) {
      v2f bf;
      bf.x = W_bc[ka * (2 * kN) + t * 16 + lr];
      bf.y = W_bc[(ka + 1) * (2 * kN) + t * 16 + lr];
      acc_bc[t] = wmma4f32(a, bf, acc_bc[t]);
    }
  }

  // Spill x_bc (B_sel | C_sel) for the scan kernel.
#pragma unroll
  for (int t = 0; t < 2; ++t)
#pragma unroll
    for (int r = 0; r < 8; ++r)
      xbc_ws[(size_t)(m0 + r + 8 * half) * (2 * kN) + t * 16 + lr] =
          acc_bc[t][r];

  // dt_low C/D-layout tile -> LDS, re-read as A fragments for GEMM2.
#pragma unroll
  for (int t = 0; t < 4; ++t)
#pragma unroll
    for (int r = 0; r < 8; ++r)
      lds_dt[w][r + 8 * half][t * 16 + lr] = acc_dt[t][r];

  __syncthreads();

  v2f afrag[16];  // all GEMM2 A fragments (K=64 -> 16 k-steps), hoisted
#pragma unroll
  for (int kk = 0; kk < 16; ++kk) {
    afrag[kk].x = lds_dt[w][lr][4 * kk + 2 * half];
    afrag[kk].y = lds_dt[w][lr][4 * kk + 2 * half + 1];
  }

  // GEMM2: delta_pre = dt_low(16x64) @ W_dtproj(64x1024), fused bias+softplus.
  for (int jt = 0; jt < kD / 16; ++jt) {
    v8f acc = zero8;
#pragma unroll
    for (int kk = 0; kk < 16; ++kk) {
      const int ka = 4 * kk + 2 * half;
      v2f bf;
      bf.x = W_dtproj[ka * kD + jt * 16 + lr];
      bf.y = W_dtproj[(ka + 1) * kD + jt * 16 + lr];
      acc = wmma4f32(afrag[kk], bf, acc);
    }
    const float bias = b_dtproj[jt * 16 + lr];
#pragma unroll
    for (int r = 0; r < 8; ++r)
      delta_ws[(size_t)(m0 + r + 8 * half) * kD + jt * 16 + lr] =
          softplus_f(acc[r] + bias);
  }
}

// ---------------------------------------------------------------------------
// Phase 2: chunked selective scan + output reduction.
// Block = 256 threads = 8 waves = 8 L-chunks; 32 lanes = 2 d-channels x N=16.
// Async-LDS double-buffered tiles of kTL timesteps feed the serial recurrence.
// ---------------------------------------------------------------------------
__global__ __launch_bounds__(256) void s6_scan(
    const float* __restrict__ x,
    const float* __restrict__ A_log,      // (D, N)
    const float* __restrict__ D_param,    // (D)
    const float* __restrict__ delta_ws,   // (M, D)
    const float* __restrict__ xbc_ws,     // (M, 32)
    float* __restrict__ y)                // (M, D)
{
  __shared__ __align__(16) float s_bc[kChunks][2][kTL][2 * kN];  // 32 KB
  __shared__ __align__(16) float s_dx[kChunks][2][kTL][4];       //  4 KB
  __shared__ float s_a[kChunks][2][kN];                          //  1 KB
  __shared__ float s_h[kChunks][2][kN];                          //  1 KB

  const int dpair = blockIdx.x % (kD / 2);
  const int b     = blockIdx.x / (kD / 2);
  const int chunk = threadIdx.x >> 5;        // wave id = L-chunk
  const int lane  = threadIdx.x & 31;
  const int sub   = (lane >> 4) & 1;         // which d channel in the wave
  const int n     = lane & 15;               // state index
  const int d0    = dpair * 2;               // even channel of the pair
  const int d     = d0 + sub;
  const int l0    = chunk * kLC;

  const float Acoef = -__expf(A_log[d * kN + n]);

  const size_t rowbase = (size_t)b * kL + l0;
  const float* bc_g = xbc_ws + rowbase * (2 * kN);   // contiguous rows
  const float* d_g  = delta_ws + rowbase * kD + d0;  // stride-kD column pair
  const float* x_g  = x + rowbase * kD + d0;

  // Issue one tile's worth of async DMA into LDS (6 instructions / tile):
  //   x_bc: kTL*32 floats contiguous -> 4x b128 across 32 lanes
  //   delta,x: lanes 0..15 each grab the (d0,d0+1) pair for one timestep (b64)
  auto stage = [&](int t) {
    const int buf = t & 1;
    const float* g = bc_g + (size_t)t * kTL * (2 * kN);
    float* lb = &s_bc[chunk][buf][0][0];
#pragma unroll
    for (int j = 0; j < 4; ++j)
      async_b128(g + j * 128 + lane * 4, lb + j * 128 + lane * 4);
    if (lane < kTL) {
      async_b64(d_g + (size_t)(t * kTL + lane) * kD, &s_dx[chunk][buf][lane][0]);
      async_b64(x_g + (size_t)(t * kTL + lane) * kD, &s_dx[chunk][buf][lane][2]);
    }
  };

  // ---- Pass 1: local scan from identity, tracking running product of A_bar.
  float h = 0.0f, a_cum = 1.0f;
  stage(0);
  for (int t = 0; t < kNT; ++t) {
    const int buf = t & 1;
    if (t + 1 < kNT) { stage(t + 1); WAIT_ASYNC(6); }
    else             { WAIT_ASYNC(0); }
#pragma unroll
    for (int l = 0; l < kTL; ++l) {
      const float dl = s_dx[chunk][buf][l][sub];
      const float xv = s_dx[chunk][buf][l][2 + sub];
      const float a  = __expf(dl * Acoef);
      const float bx = dl * xv * s_bc[chunk][buf][l][n];
      h = __builtin_fmaf(a, h, bx);
      a_cum *= a;
    }
  }
  s_a[chunk][sub][n] = a_cum;
  s_h[chunk][sub][n] = h;
  __syncthreads();

  // Compose carry-in: summaries of chunks 0..chunk-1 under (a,b) o (a',b').
  float hin = 0.0f;
  for (int c = 0; c < chunk; ++c)
    hin = __builtin_fmaf(s_a[c][sub][n], hin, s_h[c][sub][n]);

  // ---- Pass 2: rescan with carry (L2-hot), emit y.
  const float Dp = D_param[d];
  h = hin;
  stage(0);
  for (int t = 0; t < kNT; ++t) {
    const int buf = t & 1;
    if (t + 1 < kNT) { stage(t + 1); WAIT_ASYNC(6); }
    else             { WAIT_ASYNC(0); }
#pragma unroll
    for (int l = 0; l < kTL; ++l) {
      const float dl = s_dx[chunk][buf][l][sub];
      const float xv = s_dx[chunk][buf][l][2 + sub];
      const float a  = __expf(dl * Acoef);
      const float bx = dl * xv * s_bc[chunk][buf][l][n];
      h = __builtin_fmaf(a, h, bx);
      float contrib = s_bc[chunk][buf][l][kN + n] * h;  // C_sel * h
      contrib += __shfl_xor(contrib, 8, 32);            // 16-lane butterfly
      contrib += __shfl_xor(contrib, 4, 32);
      contrib += __shfl_xor(contrib, 2, 32);
      contrib += __shfl_xor(contrib, 1, 32);
      if (n == 0)
        y[(rowbase + t * kTL + l) * kD + d] = __builtin_fmaf(Dp, xv, contrib);
    }
  }
}

// ---------------------------------------------------------------------------
extern "C" void kernel_launch(void* const* d_in, const int* in_sizes, int n_in,
                              void* d_out, int out_size, void* d_ws,
                              size_t ws_size, hipStream_t stream) {
  (void)in_sizes; (void)n_in; (void)out_size; (void)ws_size;

  const float* x        = (const float*)d_in[0];
  const float* A_log    = (const float*)d_in[1];
  const float* D_param  = (const float*)d_in[2];
  const float* W_bc     = (const float*)d_in[3];
  const float* W_dt     = (const float*)d_in[4];
  const float* W_dtproj = (const float*)d_in[5];
  const float* b_dtproj = (const float*)d_in[6];
  float* y = (float*)d_out;

  float* delta_ws = (float*)d_ws;                      // 16 MB (M x D)
  float* xbc_ws   = delta_ws + (size_t)kM * kD;        // 512 KB (M x 32)

  // 256 row-tiles, 2 waves (tiles) per block.
  s6_proj<<<dim3(kM / 32), dim3(64), 0, stream>>>(
      x, W_bc, W_dt, W_dtproj, b_dtproj, delta_ws, xbc_ws);

  // One block per (b, d-pair): 2*512 = 1024 blocks, 8 waves each.
  s6_scan<<<dim3(kB * (kD / 2)), dim3(256), 0, stream>>>(
      x, A_log, D_param, delta_ws, xbc_ws, y);
}